// RegularizationLoss_68573447847948
// MI455X (gfx1250) — compile-verified
//
#include <hip/hip_runtime.h>
#include <math.h>

typedef __attribute__((ext_vector_type(2))) float v2f;
typedef __attribute__((ext_vector_type(8))) float v8f;

#define KNN 10
#define TILE_COLS 128
#define ROWS_PER_BLOCK 16
#define THREADS 256
#define DOT_STRIDE 132   // floats; 528 B per row (16B aligned), halves on disjoint bank windows
#define BIG 3.4e38f

// ---------------------------------------------------------------------------
// Kernel 0: zero the 4 accumulators (sparsity, opacity, scale, smooth)
// ---------------------------------------------------------------------------
__global__ void init_kernel(float* acc) {
  if (threadIdx.x < 4) acc[threadIdx.x] = 0.0f;
}

// ---------------------------------------------------------------------------
// Kernel 1: elementwise losses + |p|^2 precompute
// acc[0] += sum |op|   acc[1] += sum (op-0.5)^2   acc[2] += sum |scale-1|
// ---------------------------------------------------------------------------
__global__ __launch_bounds__(THREADS)
void prep_kernel(const float* __restrict__ pos,
                 const float* __restrict__ opac,
                 const float* __restrict__ scl,
                 float* __restrict__ acc,
                 float* __restrict__ sq, int N) {
  __shared__ float s_red[3][THREADS / 32];
  int i = blockIdx.x * blockDim.x + threadIdx.x;
  float v0 = 0.0f, v1 = 0.0f, v2 = 0.0f;
  if (i < N) {
    float o = opac[i];
    v0 = fabsf(o);
    float t = o - 0.5f;
    v1 = t * t;
    float x = pos[3 * i], y = pos[3 * i + 1], z = pos[3 * i + 2];
    sq[i] = x * x + y * y + z * z;
    v2 = fabsf(scl[3 * i] - 1.0f) + fabsf(scl[3 * i + 1] - 1.0f) +
         fabsf(scl[3 * i + 2] - 1.0f);
  }
  #pragma unroll
  for (int off = 16; off > 0; off >>= 1) {
    v0 += __shfl_down(v0, off, 32);
    v1 += __shfl_down(v1, off, 32);
    v2 += __shfl_down(v2, off, 32);
  }
  int lane = threadIdx.x & 31, w = threadIdx.x >> 5;
  if (lane == 0) { s_red[0][w] = v0; s_red[1][w] = v1; s_red[2][w] = v2; }
  __syncthreads();
  if (threadIdx.x == 0) {
    float a0 = 0.0f, a1 = 0.0f, a2 = 0.0f;
    #pragma unroll
    for (int k = 0; k < THREADS / 32; ++k) {
      a0 += s_red[0][k]; a1 += s_red[1][k]; a2 += s_red[2][k];
    }
    atomicAdd(&acc[0], a0);
    atomicAdd(&acc[1], a1);
    atomicAdd(&acc[2], a2);
  }
}

// ---------------------------------------------------------------------------
// Branchless sorted-insert stage: keeps list ascending, bubbles larger value.
// Pure select form: 1x v_cmp + 4x v_cndmask (pairs into v_dual_cndmask),
// avoiding the NaN-canonicalization ops that fminf/fmaxf force.
// ---------------------------------------------------------------------------
__device__ __forceinline__ void insert10(float d, int jj,
                                         float (&bd)[KNN], int (&bj)[KNN]) {
  #pragma unroll
  for (int t = 0; t < KNN; ++t) {
    bool c     = d < bd[t];
    float dmin = c ? d : bd[t];
    float dmax = c ? bd[t] : d;
    int jmin   = c ? jj : bj[t];
    int jmax   = c ? bj[t] : jj;
    bd[t] = dmin; bj[t] = jmin;
    d = dmax; jj = jmax;
  }
}

// ---------------------------------------------------------------------------
// Kernel 2: WMMA-tiled pairwise d^2 + per-row top-10 + smoothness sum.
// Producer: 8 waves x V_WMMA_F32_16X16X4_F32 build a full 16x128 d2 tile in
// LDS (diagonal pre-replaced with +BIG). Scan: thread (r,seg) reads its 8
// candidates as two ds_load_b128 and does guarded branchless insertion.
// ---------------------------------------------------------------------------
__global__ __launch_bounds__(THREADS)
void knn_smooth_kernel(const float* __restrict__ pos,
                       const float* __restrict__ opac,
                       const float* __restrict__ sq,
                       float* __restrict__ acc, int N) {
  __shared__ float s_d2[ROWS_PER_BLOCK * DOT_STRIDE];
  __shared__ float s_cand_d[ROWS_PER_BLOCK][16][KNN];
  __shared__ int   s_cand_j[ROWS_PER_BLOCK][16][KNN];

  const int tid  = threadIdx.x;
  const int lane = tid & 31;
  const int wave = tid >> 5;       // 8 waves
  const int half = lane >> 4;      // K-half selector for 16x16x4 layout
  const int l16  = lane & 15;
  const int rowBase = blockIdx.x * ROWS_PER_BLOCK;

  // A operand (constant): lanes 0-15 -> (x,y)=K0,K1 ; lanes 16-31 -> (z,0)=K2,K3
  v2f a;
  {
    int m = rowBase + l16;
    float px = pos[3 * m], py = pos[3 * m + 1], pz = pos[3 * m + 2];
    a[0] = half ? pz : px;
    a[1] = half ? 0.0f : py;
  }

  // Per-lane row norms + global row ids for the 8 C rows this lane produces
  float sq_rows[8];
  int   row_ids[8];
  #pragma unroll
  for (int rr = 0; rr < 8; ++rr) {
    row_ids[rr] = rowBase + rr + 8 * half;
    sq_rows[rr] = sq[row_ids[rr]];
  }

  // scan identity: thread (r, seg) owns row r, 8 cols per tile
  const int r   = tid >> 4;        // 0..15
  const int seg = tid & 15;        // 0..15

  float best_d[KNN];
  int   best_j[KNN];
  #pragma unroll
  for (int k = 0; k < KNN; ++k) { best_d[k] = BIG; best_j[k] = 0; }

  const int numTiles = N / TILE_COLS;
  for (int ct = 0; ct < numTiles; ++ct) {
    const int colBase = ct * TILE_COLS;

    if (ct + 1 < numTiles && tid < TILE_COLS)   // speculative next-tile prefetch
      __builtin_prefetch(&pos[3 * (colBase + TILE_COLS + tid)], 0, 1);

    // ---- producer: one 16x16 WMMA tile per wave -> d2 directly into LDS ----
    {
      int col = colBase + wave * 16 + l16;
      float qx = pos[3 * col], qy = pos[3 * col + 1], qz = pos[3 * col + 2];
      float sqc = sq[col];
      v2f b;
      b[0] = half ? qz : qx;
      b[1] = half ? 0.0f : qy;

      v8f c = {};
      c = __builtin_amdgcn_wmma_f32_16x16x4_f32(
          /*neg_a=*/false, a, /*neg_b=*/false, b,
          /*c_mod=*/(short)0, c, /*reuse_a=*/false, /*reuse_b=*/false);

      const int colLocal = wave * 16 + l16;
      #pragma unroll
      for (int rr = 0; rr < 8; ++rr) {
        float d2 = sq_rows[rr] + sqc - 2.0f * c[rr];
        d2 = (col == row_ids[rr]) ? BIG : d2;   // exclude self here, not in scan
        s_d2[(rr + 8 * half) * DOT_STRIDE + colLocal] = d2;
      }
    }
    __syncthreads();

    // ---- scan: 8 candidates via two b128 LDS loads, guarded insert ----
    {
      const float* rowPtr = &s_d2[r * DOT_STRIDE + seg * 8];
      float4 d4a = *(const float4*)(rowPtr);
      float4 d4b = *(const float4*)(rowPtr + 4);
      const int j0 = colBase + seg * 8;
      float cd[8] = {d4a.x, d4a.y, d4a.z, d4a.w, d4b.x, d4b.y, d4b.z, d4b.w};
      #pragma unroll
      for (int k = 0; k < 8; ++k) {
        if (cd[k] < best_d[KNN - 1]) insert10(cd[k], j0 + k, best_d, best_j);
      }
    }
    __syncthreads();  // protect s_d2 against next tile's writes
  }

  // dump per-thread top-10 candidates
  #pragma unroll
  for (int k = 0; k < KNN; ++k) {
    s_cand_d[r][seg][k] = best_d[k];
    s_cand_j[r][seg][k] = best_j[k];
  }
  __syncthreads();

  // 16-way merge per row: select 10 smallest of 160 candidates (cold path)
  if (tid < ROWS_PER_BLOCK) {
    const int rr = tid;
    const float op_i = opac[rowBase + rr];
    float ssum = 0.0f;
    for (int k = 0; k < KNN; ++k) {
      float bd = BIG; int bs = 0, bq = 0;
      for (int s2 = 0; s2 < 16; ++s2)
        #pragma unroll
        for (int q = 0; q < KNN; ++q) {
          float d = s_cand_d[rr][s2][q];
          if (d < bd) { bd = d; bs = s2; bq = q; }
        }
      s_cand_d[rr][bs][bq] = BIG;   // mark consumed
      ssum += fabsf(op_i - opac[s_cand_j[rr][bs][bq]]);
    }
    atomicAdd(&acc[3], ssum);
  }
}

// ---------------------------------------------------------------------------
// Kernel 3: combine weighted means
// ---------------------------------------------------------------------------
__global__ void finalize_kernel(const float* __restrict__ acc,
                                float* __restrict__ out, int N) {
  if (threadIdx.x == 0) {
    float fN = (float)N;
    float sparsity = acc[0] / fN;
    float opacity  = acc[1] / fN;
    float scale    = acc[2] / (3.0f * fN);
    float smooth   = acc[3] / (fN * (float)KNN);
    out[0] = 0.01f * sparsity + 0.1f * smooth + scale + opacity;
  }
}

// ---------------------------------------------------------------------------
extern "C" void kernel_launch(void* const* d_in, const int* in_sizes, int n_in,
                              void* d_out, int out_size, void* d_ws, size_t ws_size,
                              hipStream_t stream) {
  const float* pos  = (const float*)d_in[0];   // (N,3) f32
  const float* opac = (const float*)d_in[1];   // (N,)  f32
  const float* scl  = (const float*)d_in[2];   // (N,3) f32
  float* out = (float*)d_out;                  // scalar f32
  const int N = in_sizes[1];                   // 16384 (multiple of 128)

  float* acc = (float*)d_ws;                   // 4 accumulators
  float* sq  = acc + 4;                        // N floats of |p|^2

  init_kernel<<<1, 32, 0, stream>>>(acc);
  prep_kernel<<<(N + THREADS - 1) / THREADS, THREADS, 0, stream>>>(
      pos, opac, scl, acc, sq, N);
  knn_smooth_kernel<<<N / ROWS_PER_BLOCK, THREADS, 0, stream>>>(
      pos, opac, sq, acc, N);
  finalize_kernel<<<1, 32, 0, stream>>>(acc, out, N);
}